// PillarNet_63977832841497
// MI455X (gfx1250) — compile-verified
//
#include <hip/hip_runtime.h>
#include <stdint.h>

// ---------------- constants ----------------
#define KTOT        (1u << 20)      // 4 batches * 512 * 512 pillar keys
#define TILE_PTS    256
#define TILE_BYTES  (TILE_PTS * 24) // 6 floats per point (input)
#define FEAT_BYTES  (TILE_PTS * 40) // 10 floats per point (output) = 10240
#define SCAN_TPB    256
#define SCAN_ITEMS  16              // 256 blocks * 256 thr * 16 = 2^20

// workspace layout in 32-bit words
#define W_CNT   0u
#define W_SX    (1u << 20)
#define W_SY    (2u << 20)
#define W_SZ    (3u << 20)
#define W_RANK  (4u << 20)
#define W_BSUM  (5u << 20)            // 256 block sums
#define W_KEYS  ((5u << 20) + 256u)   // N keys

// ---------------- kernel 1: init ----------------
__global__ void pn_init(unsigned* __restrict__ ws, int* __restrict__ unq_out,
                        long long* __restrict__ grid_hw, int npts) {
    long long stride = (long long)gridDim.x * blockDim.x;
    long long tid0   = (long long)blockIdx.x * blockDim.x + threadIdx.x;
    // zero cnt + 3 sum planes (4M words)
    for (long long i = tid0; i < (long long)(4u << 20); i += stride) ws[i] = 0u;
    // pre-fill unq rows with -1 (padding semantics of jnp.unique(size=n, fill=-1))
    long long nunq = (long long)npts * 3;
    for (long long i = tid0; i < nunq; i += stride) unq_out[i] = -1;
    if (tid0 == 0) { grid_hw[0] = 512; grid_hw[1] = 512; }
}

// ---------------- kernel 2: accumulate (async global->LDS staged) ----------------
__global__ void pn_accum(const float* __restrict__ pts, unsigned* __restrict__ ws,
                         int npts, int ntiles) {
    __shared__ float sbuf[2][TILE_PTS * 6];   // 12 KB double buffer, image of tile bytes

    unsigned* cnt  = ws + W_CNT;
    float*    sx   = (float*)(ws + W_SX);
    float*    sy   = (float*)(ws + W_SY);
    float*    sz   = (float*)(ws + W_SZ);
    unsigned* keys = ws + W_KEYS;

    const char*     gbase   = (const char*)pts;
    const long long max_off = (long long)npts * 24 - 8;  // clamp target for tail tile
    const int       tid     = threadIdx.x;

    // each thread DMAs 3 x 8B chunks of the 6144B tile into LDS (ASYNCcnt-tracked)
    auto issue = [&](int tile, int buf) {
        unsigned  lbase = (unsigned)(uintptr_t)(&sbuf[buf][0]); // low 32b of generic = LDS offset
        long long tb    = (long long)tile * TILE_BYTES + (long long)tid * 8;
#pragma unroll
        for (int k = 0; k < 3; ++k) {
            long long go = tb + (long long)k * 2048;
            if (go > max_off) go = max_off;               // safe dummy source for tail
            const char* ga = gbase + go;
            unsigned    la = lbase + (unsigned)(tid * 8 + k * 2048);
            asm volatile("global_load_async_to_lds_b64 %0, %1, off"
                         :: "v"(la), "v"(ga) : "memory");
        }
    };

    int cur  = 0;
    int tile = blockIdx.x;
    if (tile < ntiles) issue(tile, 0);

    for (; tile < ntiles; tile += gridDim.x) {
        int nxt = tile + gridDim.x;                       // block-uniform branch
        if (nxt < ntiles) {
            issue(nxt, cur ^ 1);                          // prefetch next tile
            // async loads complete in order: asynccnt<=3 => current tile's 3 done
            asm volatile("s_wait_asynccnt 0x3" ::: "memory");
        } else {
            asm volatile("s_wait_asynccnt 0x0" ::: "memory");
        }
        __syncthreads();

        int i = tile * TILE_PTS + tid;
        if (i < npts) {
            const float* p = &sbuf[cur][tid * 6];
            float x = p[1], y = p[2], z = p[3];
            int   bi = (int)p[0];
            int   cx = (int)((x + 51.2f) / 0.2f);
            int   cy = (int)((y + 51.2f) / 0.2f);
            unsigned key = ((((unsigned)bi << 9) | (unsigned)cx) << 9) | (unsigned)cy;
            keys[i] = key;
            atomicAdd(&cnt[key], 1u);                      // L2-resident (16MB grid << 192MB L2)
            atomicAdd(&sx[key], x);
            atomicAdd(&sy[key], y);
            atomicAdd(&sz[key], z);
        }
        __syncthreads();
        cur ^= 1;
    }
}

// ---------------- kernel 3a: per-block scan of occupancy flags ----------------
__global__ void pn_scanA(unsigned* __restrict__ ws) {
    unsigned* cnt  = ws + W_CNT;
    unsigned* rank = ws + W_RANK;
    unsigned* bsum = ws + W_BSUM;
    __shared__ unsigned ssum[SCAN_TPB];

    int tid  = threadIdx.x;
    int base = blockIdx.x * (SCAN_TPB * SCAN_ITEMS) + tid * SCAN_ITEMS;

    unsigned f[SCAN_ITEMS], tsum = 0;
#pragma unroll
    for (int j = 0; j < SCAN_ITEMS; ++j) { f[j] = cnt[base + j] ? 1u : 0u; tsum += f[j]; }

    ssum[tid] = tsum;
    __syncthreads();
    for (int s = 1; s < SCAN_TPB; s <<= 1) {              // Hillis-Steele inclusive
        unsigned v = (tid >= s) ? ssum[tid - s] : 0u;
        __syncthreads();
        ssum[tid] += v;
        __syncthreads();
    }
    unsigned texcl = ssum[tid] - tsum;

    unsigned run = 0;
#pragma unroll
    for (int j = 0; j < SCAN_ITEMS; ++j) { rank[base + j] = texcl + run; run += f[j]; }

    if (tid == SCAN_TPB - 1) bsum[blockIdx.x] = ssum[tid]; // block total
}

// ---------------- kernel 3b: scan the 256 block sums ----------------
__global__ void pn_scanB(unsigned* __restrict__ ws) {
    unsigned* bsum = ws + W_BSUM;
    __shared__ unsigned s[SCAN_TPB];
    int tid = threadIdx.x;
    unsigned v = bsum[tid];
    s[tid] = v;
    __syncthreads();
    for (int st = 1; st < SCAN_TPB; st <<= 1) {
        unsigned u = (tid >= st) ? s[tid - st] : 0u;
        __syncthreads();
        s[tid] += u;
        __syncthreads();
    }
    bsum[tid] = s[tid] - v;                                // exclusive block offsets
}

// ---------------- kernel 3c: fixup ranks + scatter sorted unq rows ----------------
__global__ void pn_scanC(unsigned* __restrict__ ws, int* __restrict__ unq_out) {
    unsigned* cnt  = ws + W_CNT;
    unsigned* rank = ws + W_RANK;
    unsigned* bsum = ws + W_BSUM;

    int      base = blockIdx.x * (SCAN_TPB * SCAN_ITEMS) + threadIdx.x * SCAN_ITEMS;
    unsigned off  = bsum[blockIdx.x];
#pragma unroll
    for (int j = 0; j < SCAN_ITEMS; ++j) {
        unsigned i = (unsigned)(base + j);
        unsigned r = rank[i] + off;
        rank[i] = r;                                       // final sorted rank of key i
        if (cnt[i]) {                                      // occupied -> sorted unq row [b, y, x]
            int b  = (int)(i >> 18);
            int cx = (int)((i >> 9) & 511u);
            int cy = (int)(i & 511u);
            unq_out[3u * r + 0u] = b;
            unq_out[3u * r + 1u] = cy;
            unq_out[3u * r + 2u] = cx;
        }
    }
}

// ---------------- kernel 4: finalize features (async LDS->global stores) ----------------
__global__ void pn_final(const float* __restrict__ pts, const unsigned* __restrict__ ws,
                         float* __restrict__ feat, int* __restrict__ inv_out,
                         int npts, int ntiles) {
    __shared__ float sfeat[2][TILE_PTS * 10];  // 20 KB double buffer, image of output tile

    const unsigned* cnt  = ws + W_CNT;
    const float*    sx   = (const float*)(ws + W_SX);
    const float*    sy   = (const float*)(ws + W_SY);
    const float*    sz   = (const float*)(ws + W_SZ);
    const unsigned* rank = ws + W_RANK;
    const unsigned* keys = ws + W_KEYS;

    const int tid = threadIdx.x;
    int cur = 0;

    for (int tile = blockIdx.x; tile < ntiles; tile += gridDim.x) {
        long long i    = (long long)tile * TILE_PTS + tid;
        bool      full = ((long long)tile * TILE_PTS + TILE_PTS) <= (long long)npts; // uniform

        // ensure stores issued 2 iterations ago (same buffer) are done before overwrite:
        // each wave bounds its own counter, the barrier extends it block-wide
        asm volatile("s_wait_asynccnt 0x5" ::: "memory");
        __syncthreads();

        float o[10];
        if (i < npts) {
            unsigned key = keys[i];
            unsigned r   = rank[key];
            inv_out[i]   = (int)r;

            float c  = fmaxf((float)cnt[key], 1.0f);
            float mx = sx[key] / c, my = sy[key] / c, mz = sz[key] / c;

            const float* p = pts + i * 6;
            float x = p[1], y = p[2], z = p[3], f1 = p[4], f2 = p[5];
            int   cx = (int)((x + 51.2f) / 0.2f);
            int   cy = (int)((y + 51.2f) / 0.2f);
            float pcx = (float)cx * 0.2f + 0.1f - 51.2f;
            float pcy = (float)cy * 0.2f + 0.1f - 51.2f;

            o[0] = x;  o[1] = y;  o[2] = z;  o[3] = f1; o[4] = f2;
            o[5] = x - mx; o[6] = y - my; o[7] = z - mz;
            o[8] = x - pcx; o[9] = y - pcy;
        }

        if (full) {
            // stage row in LDS (stride-10 floats: conflict-free over 32 lanes)
#pragma unroll
            for (int j = 0; j < 10; ++j) sfeat[cur][tid * 10 + j] = o[j];
            __syncthreads();   // all rows staged before DMA reads them

            // drain 10240B tile: 5 x 8B sequential chunks per thread, ASYNCcnt-tracked
            unsigned    lbase = (unsigned)(uintptr_t)(&sfeat[cur][0]);
            const char* gtile = (const char*)feat + (long long)tile * FEAT_BYTES;
#pragma unroll
            for (int k = 0; k < 5; ++k) {
                unsigned    la = lbase + (unsigned)(tid * 8 + k * 2048);
                const char* ga = gtile + (tid * 8 + k * 2048);
                asm volatile("global_store_async_from_lds_b64 %0, %1, off"
                             :: "v"(ga), "v"(la) : "memory");
            }
            cur ^= 1;
        } else if (i < npts) {
            // partial tail tile: guarded direct stores (no address clamping on writes)
            float* op = feat + i * 10;
#pragma unroll
            for (int j = 0; j < 10; ++j) op[j] = o[j];
        }
    }
    asm volatile("s_wait_asynccnt 0x0" ::: "memory");  // drain before end (S_ENDPGM also waits)
}

// ---------------- launcher ----------------
extern "C" void kernel_launch(void* const* d_in, const int* in_sizes, int n_in,
                              void* d_out, int out_size, void* d_ws, size_t ws_size,
                              hipStream_t stream) {
    const float* pts  = (const float*)d_in[0];
    int          npts = in_sizes[0] / 6;

    unsigned* ws = (unsigned*)d_ws;

    // output layout (flat, return order): features[N*10] f32, unq[N*3] i32,
    // unq_inv[N] i32, grid_hw[2] i64
    float*     feat    = (float*)d_out;
    int*       unq_out = (int*)d_out + (long long)npts * 10;
    int*       inv_out = unq_out + (long long)npts * 3;
    long long* grid_hw = (long long*)(inv_out + npts);

    int ntiles = (npts + TILE_PTS - 1) / TILE_PTS;
    int ablk   = ntiles < 2048 ? ntiles : 2048;

    pn_init <<<2048, 256, 0, stream>>>(ws, unq_out, grid_hw, npts);
    pn_accum<<<ablk, TILE_PTS, 0, stream>>>(pts, ws, npts, ntiles);
    pn_scanA<<<256, SCAN_TPB, 0, stream>>>(ws);
    pn_scanB<<<1,   SCAN_TPB, 0, stream>>>(ws);
    pn_scanC<<<256, SCAN_TPB, 0, stream>>>(ws, unq_out);
    pn_final<<<ablk, TILE_PTS, 0, stream>>>(pts, ws, feat, inv_out, npts, ntiles);
}